// SimpleGCN_13554916786416
// MI455X (gfx1250) — compile-verified
//
#include <hip/hip_runtime.h>
#include <hip/hip_bf16.h>

typedef float v2f __attribute__((ext_vector_type(2)));
typedef float v8f __attribute__((ext_vector_type(8)));

#define HID 128

// ---------------- utility ----------------
__global__ void zero_kernel(float* __restrict__ p, long long n) {
  long long i = (long long)blockIdx.x * blockDim.x + threadIdx.x;
  if (i < n) p[i] = 0.0f;
}

// ---------------- degree ----------------
__global__ void deg_kernel(const int* __restrict__ ei, float* __restrict__ deg, long long E) {
  long long e = (long long)blockIdx.x * blockDim.x + threadIdx.x;
  if (e < E) {
    int dst = ei[E + e];
    unsafeAtomicAdd(&deg[dst], 1.0f);
  }
}

__global__ void dinv_kernel(float* __restrict__ deg, long long N) {
  long long i = (long long)blockIdx.x * blockDim.x + threadIdx.x;
  if (i < N) deg[i] = rsqrtf(deg[i] + 1.0f);  // deg includes self-loop
}

// ---------------- fp32 WMMA GEMM: Y[N,128] = X[N,128] @ W[128,128] ----------------
// One wave computes a 16-row block across all 128 output columns.
// W is staged in LDS (64 KB) once per 256-thread block.
__global__ __launch_bounds__(256) void gemm_kernel(const float* __restrict__ X,
                                                   const float* __restrict__ W,
                                                   float* __restrict__ Y,
                                                   int nrows) {
  __shared__ float Wl[HID * HID];  // 64 KB
  const int t = threadIdx.x;

  // cooperative load of W into LDS, float4 vectorized
  const float4* Wv = (const float4*)W;
  float4* Wlv = (float4*)Wl;
  #pragma unroll 4
  for (int i = t; i < HID * HID / 4; i += 256) Wlv[i] = Wv[i];
  __syncthreads();

  const int wave = t >> 5;          // 0..7
  const int lane = t & 31;
  const int rowBlock = blockIdx.x * 8 + wave;  // 16-row tile index
  if (rowBlock * 16 >= nrows) return;          // wave-uniform exit

  const int m  = lane & 15;   // row (A) / col (B,C) within tile
  const int kh = lane >> 4;   // K-half selector per ISA 32-bit A 16x4 layout

  const float* xrow = X + (long long)(rowBlock * 16 + m) * HID;

  v8f c[8];
  #pragma unroll
  for (int nt = 0; nt < 8; ++nt) c[nt] = (v8f){0.f,0.f,0.f,0.f,0.f,0.f,0.f,0.f};

  for (int k0 = 0; k0 < HID; k0 += 4) {
    // A fragment: lanes 0-15 -> K = k0,k0+1 ; lanes 16-31 -> K = k0+2,k0+3
    v2f a;
    a.x = xrow[k0 + 2 * kh + 0];
    a.y = xrow[k0 + 2 * kh + 1];
    #pragma unroll
    for (int nt = 0; nt < 8; ++nt) {
      const int n = nt * 16 + m;
      v2f b;
      b.x = Wl[(k0 + 2 * kh + 0) * HID + n];
      b.y = Wl[(k0 + 2 * kh + 1) * HID + n];
      c[nt] = __builtin_amdgcn_wmma_f32_16x16x4_f32(
          /*neg_a=*/false, a, /*neg_b=*/false, b,
          /*c_mod=*/(short)0, c[nt], /*reuse_a=*/false, /*reuse_b=*/false);
    }
  }

  // C/D layout: VGPR v -> row v (lanes 0-15) or row v+8 (lanes 16-31)
  #pragma unroll
  for (int nt = 0; nt < 8; ++nt) {
    const int n = nt * 16 + m;
    float* yb = Y + (long long)(rowBlock * 16 + kh * 8) * HID + n;
    #pragma unroll
    for (int v = 0; v < 8; ++v) yb[(long long)v * HID] = c[nt][v];
  }
}

// ---------------- edge scatter: agg[dst] += xw[src] * dinv[src]*dinv[dst] ----------------
// One wave per edge; each lane handles one float4 (4 features).
__global__ void scatter_kernel(const int* __restrict__ ei,
                               const float* __restrict__ xw,
                               const float* __restrict__ dinv,
                               float* __restrict__ agg,
                               long long E) {
  long long tid = (long long)blockIdx.x * blockDim.x + threadIdx.x;
  long long e = tid >> 5;          // wave-uniform (blockDim multiple of 32)
  if (e >= E) return;
  const int f4 = (int)(tid & 31);
  const int src = ei[e];
  const int dst = ei[E + e];
  const float norm = dinv[src] * dinv[dst];
  const float4 v = ((const float4*)(xw + (long long)src * HID))[f4];
  float* base = agg + (long long)dst * HID + f4 * 4;
  unsafeAtomicAdd(base + 0, v.x * norm);
  unsafeAtomicAdd(base + 1, v.y * norm);
  unsafeAtomicAdd(base + 2, v.z * norm);
  unsafeAtomicAdd(base + 3, v.w * norm);
}

// ---------------- combine layer1: h = relu(agg + xw*dinv^2 + b) (in place into agg) ----------------
__global__ void combine_relu_kernel(float* __restrict__ agg, const float* __restrict__ xw,
                                    const float* __restrict__ dinv,
                                    const float* __restrict__ bias, long long total) {
  long long i = (long long)blockIdx.x * blockDim.x + threadIdx.x;
  if (i >= total) return;
  const long long n = i >> 7;
  const int f = (int)(i & (HID - 1));
  const float di = dinv[n];
  const float v = agg[i] + xw[i] * di * di + bias[f];
  agg[i] = fmaxf(v, 0.0f);
}

// ---------------- combine layer2 fused with mean-pool accumulation ----------------
__global__ void combine_pool_kernel(const float* __restrict__ agg, const float* __restrict__ xw,
                                    const float* __restrict__ dinv,
                                    const float* __restrict__ bias,
                                    const int* __restrict__ batch,
                                    float* __restrict__ pool, long long total) {
  long long i = (long long)blockIdx.x * blockDim.x + threadIdx.x;
  if (i >= total) return;
  const long long n = i >> 7;
  const int f = (int)(i & (HID - 1));
  const float di = dinv[n];
  const float v = agg[i] + xw[i] * di * di + bias[f];
  unsafeAtomicAdd(&pool[(long long)batch[n] * HID + f], v);
}

__global__ void count_kernel(const int* __restrict__ batch, float* __restrict__ cnt, long long N) {
  long long i = (long long)blockIdx.x * blockDim.x + threadIdx.x;
  if (i < N) unsafeAtomicAdd(&cnt[batch[i]], 1.0f);
}

// ---------------- final: out[g] = (pool[g] . Wlin)/max(cnt,1) + blin ----------------
__global__ void final_kernel(const float* __restrict__ pool, const float* __restrict__ cnt,
                             const float* __restrict__ Wlin, const float* __restrict__ blin,
                             float* __restrict__ out, int G) {
  const int wave = threadIdx.x >> 5;
  const int lane = threadIdx.x & 31;
  const int g = blockIdx.x * (blockDim.x >> 5) + wave;
  if (g >= G) return;
  float s = 0.0f;
  #pragma unroll
  for (int f = lane; f < HID; f += 32) s += pool[(long long)g * HID + f] * Wlin[f];
  #pragma unroll
  for (int off = 16; off > 0; off >>= 1) s += __shfl_down(s, off, 32);
  if (lane == 0) out[g] = s / fmaxf(cnt[g], 1.0f) + blin[0];
}

// ---------------- host side ----------------
static inline long long cdiv(long long a, long long b) { return (a + b - 1) / b; }
static inline size_t alignup(size_t x) { return (x + 255) & ~(size_t)255; }

extern "C" void kernel_launch(void* const* d_in, const int* in_sizes, int n_in,
                              void* d_out, int out_size, void* d_ws, size_t ws_size,
                              hipStream_t stream) {
  const float* x     = (const float*)d_in[0];
  const int*   ei    = (const int*)d_in[1];
  const int*   batch = (const int*)d_in[2];
  const float* W1    = (const float*)d_in[3];
  const float* b1    = (const float*)d_in[4];
  const float* W2    = (const float*)d_in[5];
  const float* b2    = (const float*)d_in[6];
  const float* Wlin  = (const float*)d_in[7];
  const float* blin  = (const float*)d_in[8];
  float* out = (float*)d_out;

  const long long N = (long long)in_sizes[0] / HID;   // 100000
  const long long E = (long long)in_sizes[1] / 2;     // 1600000
  const int G = out_size;                             // 2048 (OUT=1)
  const long long NH = N * HID;

  // workspace layout
  char* ws = (char*)d_ws;
  float* xw   = (float*)ws;                       ws += alignup(NH * sizeof(float));
  float* agg  = (float*)ws;                       ws += alignup(NH * sizeof(float));
  float* dinv = (float*)ws;                       ws += alignup(N * sizeof(float));
  float* pool = (float*)ws;                       ws += alignup((size_t)G * HID * sizeof(float));
  float* cnt  = (float*)ws;

  const int B = 256;
  const int gN   = (int)cdiv(N, B);
  const int gNH  = (int)cdiv(NH, B);
  const int gE   = (int)cdiv(E, B);
  const int gE32 = (int)cdiv(E * 32, B);
  const int gGemm = (int)cdiv(cdiv(N, 16), 8);    // 8 waves (16-row tiles) per block

  // degrees -> dinv (in place)
  zero_kernel<<<gN, B, 0, stream>>>(dinv, N);
  deg_kernel<<<gE, B, 0, stream>>>(ei, dinv, E);
  dinv_kernel<<<gN, B, 0, stream>>>(dinv, N);

  // ---- layer 1 ----
  gemm_kernel<<<gGemm, B, 0, stream>>>(x, W1, xw, (int)N);
  zero_kernel<<<gNH, B, 0, stream>>>(agg, NH);
  scatter_kernel<<<gE32, B, 0, stream>>>(ei, xw, dinv, agg, E);
  combine_relu_kernel<<<gNH, B, 0, stream>>>(agg, xw, dinv, b1, NH);  // h1 -> agg

  // ---- layer 2 ----
  gemm_kernel<<<gGemm, B, 0, stream>>>(agg, W2, xw, (int)N);          // xw2 -> xw
  zero_kernel<<<gNH, B, 0, stream>>>(agg, NH);                        // reuse agg
  scatter_kernel<<<gE32, B, 0, stream>>>(ei, xw, dinv, agg, E);

  // ---- pool + linear ----
  zero_kernel<<<(int)cdiv((long long)G * HID, B), B, 0, stream>>>(pool, (long long)G * HID);
  zero_kernel<<<(int)cdiv(G, B), B, 0, stream>>>(cnt, G);
  count_kernel<<<gN, B, 0, stream>>>(batch, cnt, N);
  combine_pool_kernel<<<gNH, B, 0, stream>>>(agg, xw, dinv, b2, batch, pool, NH);
  final_kernel<<<(int)cdiv(G, B / 32), B, 0, stream>>>(pool, cnt, Wlin, blin, out, G);
}